// GraphFeatureExtractor_12695923327318
// MI455X (gfx1250) — compile-verified
//
#include <hip/hip_runtime.h>
#include <hip/hip_bf16.h>

typedef float v2f __attribute__((ext_vector_type(2)));
typedef float v8f __attribute__((ext_vector_type(8)));

#define DIST_THRESH 150.0f
#define ROW_EPS 1e-6f
#define BN_EPS 1e-5f

// ---------------------------------------------------------------------------
// Kernel 1: bulk passthrough copy of person_features -> out (128 MB), float4.
// ---------------------------------------------------------------------------
__global__ void gfe_copy_f4(const float4* __restrict__ src, float4* __restrict__ dst) {
    long long i = (long long)blockIdx.x * blockDim.x + threadIdx.x;
    dst[i] = src[i];
}

// ---------------------------------------------------------------------------
// Kernel 2: build row-normalized adjacency A_norm[b][n][m]. 64 blocks x 64 thr.
// ---------------------------------------------------------------------------
__global__ void gfe_adjacency(const float* __restrict__ bboxes, float* __restrict__ an) {
    __shared__ float cx[64], cy[64];
    int b = blockIdx.x;
    int n = threadIdx.x;                      // 0..63
    const float* bb = bboxes + ((long long)b * 64 + n) * 4;
    cx[n] = bb[0] + 0.5f * bb[2];
    cy[n] = bb[1] + 0.5f * bb[3];
    __syncthreads();
    float myx = cx[n], myy = cy[n];
    float rowsum = 0.0f;
    for (int m = 0; m < 64; ++m) {
        float dx = myx - cx[m], dy = myy - cy[m];
        float d  = sqrtf(dx * dx + dy * dy);
        float a  = (m == n) ? 1.0f : ((d < DIST_THRESH) ? 1.0f : 0.0f);
        rowsum += a;
    }
    float inv = 1.0f / (rowsum + ROW_EPS);
    float* dst = an + ((long long)b * 64 + n) * 64;
    for (int m = 0; m < 64; ++m) {
        float dx = myx - cx[m], dy = myy - cy[m];
        float d  = sqrtf(dx * dx + dy * dy);
        float a  = (m == n) ? 1.0f : ((d < DIST_THRESH) ? 1.0f : 0.0f);
        dst[m] = a * inv;
    }
}

// ---------------------------------------------------------------------------
// Kernel 3: M1[b] = X[b]^T (512x64) * A_norm[b] (64x64), f32 WMMA 16x16x4.
// X[b][n][c] accessed via base + b*xBatchStride + n*xRowStride + c.
// Grid: (B * 32) blocks of 128 threads; wave w handles N-tile w (4 tiles).
// ---------------------------------------------------------------------------
__global__ void gfe_xt_a_wmma(const float* __restrict__ x, long long xBatchStride,
                              int xRowStride, const float* __restrict__ an,
                              float* __restrict__ m1) {
    int lane = threadIdx.x & 31;
    int wave = threadIdx.x >> 5;              // 0..3 -> N tile
    int mt   = blockIdx.x & 31;               // 0..31 -> M tile (channel rows)
    int b    = blockIdx.x >> 5;               // batch
    const float* xb = x  + (long long)b * xBatchStride;
    const float* ab = an + (long long)b * 4096;

    int mrow = mt * 16 + (lane & 15);         // channel c (A-matrix row)
    int ncol = wave * 16 + (lane & 15);       // output column m (B-matrix col)
    int koff = (lane >> 4) << 1;              // lanes 0-15: K+0/1, lanes 16-31: K+2/3

    v8f acc = {};
    for (int k = 0; k < 64; k += 4) {
        int kk = k + koff;
        v2f a, bm;
        a.x  = xb[(long long)kk * xRowStride + mrow];        // X^T[c][n=kk]
        a.y  = xb[(long long)(kk + 1) * xRowStride + mrow];
        bm.x = ab[kk * 64 + ncol];                           // A_norm[n=kk][m]
        bm.y = ab[(kk + 1) * 64 + ncol];
        acc = __builtin_amdgcn_wmma_f32_16x16x4_f32(
            false, a, false, bm, (short)0, acc, false, false);
    }
    // D 16x16 layout: VGPR r -> rows r (lanes 0-15) and r+8 (lanes 16-31)
    float* dst = m1 + ((long long)b * 512 + mt * 16) * 64;
    int rbase = (lane >> 4) << 3;
#pragma unroll
    for (int r = 0; r < 8; ++r)
        dst[(rbase + r) * 64 + ncol] = acc[r];
}

// ---------------------------------------------------------------------------
// Kernel 4: Z[b] = W (512x512) * M1[b] (512x64) + bias, f32 WMMA, K=512.
// A-operand (W) K-pair is contiguous in memory -> single 8-byte v2f load.
// ---------------------------------------------------------------------------
__global__ void gfe_wy_wmma(const float* __restrict__ w, const float* __restrict__ m1,
                            const float* __restrict__ bias, float* __restrict__ z) {
    int lane = threadIdx.x & 31;
    int wave = threadIdx.x >> 5;              // 0..3 -> N tile
    int mt   = blockIdx.x & 31;               // 0..31 -> output-channel tile
    int b    = blockIdx.x >> 5;               // batch
    const float* mb = m1 + (long long)b * 32768;

    int orow = mt * 16 + (lane & 15);         // output channel o (A-matrix row)
    int ncol = wave * 16 + (lane & 15);       // node column m
    int koff = (lane >> 4) << 1;

    const float* wrow = w + (long long)orow * 512;

    v8f acc = {};
    for (int k = 0; k < 512; k += 4) {
        int kk = k + koff;
        v2f a = *(const v2f*)(wrow + kk);                    // W[o][kk], W[o][kk+1]
        v2f bm;
        bm.x = mb[kk * 64 + ncol];                           // M1[c=kk][m]
        bm.y = mb[(kk + 1) * 64 + ncol];
        acc = __builtin_amdgcn_wmma_f32_16x16x4_f32(
            false, a, false, bm, (short)0, acc, false, false);
    }
    float* dst = z + ((long long)b * 512 + mt * 16) * 64;
    int rbase = (lane >> 4) << 3;
#pragma unroll
    for (int r = 0; r < 8; ++r) {
        int row = rbase + r;
        dst[row * 64 + ncol] = acc[r] + bias[mt * 16 + row];
    }
}

// ---------------------------------------------------------------------------
// Kernel 5: BN stats per channel o over (b, m): 512 blocks x 128 threads.
// ---------------------------------------------------------------------------
__global__ void gfe_bn_stats(const float* __restrict__ z, float* __restrict__ mean,
                             float* __restrict__ var) {
    int o = blockIdx.x;                       // 0..511
    int tid = threadIdx.x;                    // 0..127
    float s = 0.0f, ss = 0.0f;
    for (int idx = tid; idx < 4096; idx += 128) {
        int b = idx >> 6, m = idx & 63;
        float v = z[((long long)b * 512 + o) * 64 + m];
        s += v; ss += v * v;
    }
    __shared__ float sh_s[128], sh_ss[128];
    sh_s[tid] = s; sh_ss[tid] = ss;
    __syncthreads();
    for (int off = 64; off > 0; off >>= 1) {
        if (tid < off) { sh_s[tid] += sh_s[tid + off]; sh_ss[tid] += sh_ss[tid + off]; }
        __syncthreads();
    }
    if (tid == 0) {
        float mu = sh_s[0] * (1.0f / 4096.0f);
        mean[o] = mu;
        var[o]  = sh_ss[0] * (1.0f / 4096.0f) - mu * mu;
    }
}

// ---------------------------------------------------------------------------
// Kernel 6: normalize + affine + relu, transpose to (B, N, D).
// out index = (b*64 + m) * outPitch + outOffset + o
//   layer-1 intermediate h: outPitch=512, outOffset=0
//   final output slice:     outPitch=16*512, outOffset=15*512
// ---------------------------------------------------------------------------
__global__ void gfe_bn_apply(const float* __restrict__ z, const float* __restrict__ mean,
                             const float* __restrict__ var, const float* __restrict__ g,
                             const float* __restrict__ beta, float* __restrict__ out,
                             int outPitch, int outOffset) {
    long long gid = (long long)blockIdx.x * blockDim.x + threadIdx.x; // (b, m, o), o fastest
    int o = (int)(gid & 511);
    int m = (int)((gid >> 9) & 63);
    int b = (int)(gid >> 15);
    float v   = z[((long long)b * 512 + o) * 64 + m];
    float inv = rsqrtf(var[o] + BN_EPS);
    float y   = (v - mean[o]) * inv * g[o] + beta[o];
    y = fmaxf(y, 0.0f);
    out[(long long)(b * 64 + m) * outPitch + outOffset + o] = y;
}

// ---------------------------------------------------------------------------
// Launch
// ---------------------------------------------------------------------------
extern "C" void kernel_launch(void* const* d_in, const int* in_sizes, int n_in,
                              void* d_out, int out_size, void* d_ws, size_t ws_size,
                              hipStream_t stream) {
    const float* pf     = (const float*)d_in[0]; // (64,64,16,512)
    const float* bboxes = (const float*)d_in[1]; // (64,64,4)
    const float* w1     = (const float*)d_in[2];
    const float* b1     = (const float*)d_in[3];
    const float* g1     = (const float*)d_in[4];
    const float* beta1  = (const float*)d_in[5];
    const float* w2     = (const float*)d_in[6];
    const float* b2     = (const float*)d_in[7];
    const float* g2     = (const float*)d_in[8];
    const float* beta2  = (const float*)d_in[9];
    float* out = (float*)d_out;

    float* ws    = (float*)d_ws;
    float* anorm = ws;                      //  64*64*64   = 262144
    float* m1    = anorm + 262144;          //  64*512*64  = 2097152
    float* z     = m1    + 2097152;         //  64*512*64  = 2097152
    float* h     = z     + 2097152;         //  64*512*64  = 2097152
    float* mean  = h     + 2097152;         //  512
    float* var   = mean  + 512;             //  512

    // 1) passthrough copy: 64*64*16*512 floats = 8388608 float4s
    gfe_copy_f4<<<32768, 256, 0, stream>>>((const float4*)pf, (float4*)out);

    // 2) adjacency
    gfe_adjacency<<<64, 64, 0, stream>>>(bboxes, anorm);

    // ---- layer 1: X from person_features last timestep ----
    gfe_xt_a_wmma<<<64 * 32, 128, 0, stream>>>(pf + 15 * 512,
                                               (long long)64 * 16 * 512, 16 * 512,
                                               anorm, m1);
    gfe_wy_wmma<<<64 * 32, 128, 0, stream>>>(w1, m1, b1, z);
    gfe_bn_stats<<<512, 128, 0, stream>>>(z, mean, var);
    gfe_bn_apply<<<8192, 256, 0, stream>>>(z, mean, var, g1, beta1, h, 512, 0);

    // ---- layer 2: X = h (B,N,D contiguous) ----
    gfe_xt_a_wmma<<<64 * 32, 128, 0, stream>>>(h,
                                               (long long)64 * 512, 512,
                                               anorm, m1);
    gfe_wy_wmma<<<64 * 32, 128, 0, stream>>>(w2, m1, b2, z);
    gfe_bn_stats<<<512, 128, 0, stream>>>(z, mean, var);
    gfe_bn_apply<<<8192, 256, 0, stream>>>(z, mean, var, g2, beta2,
                                           out, 16 * 512, 15 * 512);
}